// GNN_41128606826950
// MI455X (gfx1250) — compile-verified
//
#include <hip/hip_runtime.h>
#include <hip/hip_bf16.h>

// GCN (3-layer, GCNConv w/ self loops + symmetric norm) for MI455X / gfx1250.
// - GEMMs: V_WMMA_F32_16X16X4_F32 (exact fp32). Weights staged in LDS in
//   fragment layout (K-pairs contiguous) -> one ds_load_b64 per B fragment.
//   A fragments: each lane loads exactly its float2 (no selects, no waste).
//   Self-loop term (agg = dinv^2 * h) fused into the GEMM epilogue.
// - Edge phase: L2-resident float4 gather + float atomics.

constexpr int kNodes = 100000;   // 6250 * 16 rows -> exact 16-row tiling
constexpr int kEdges = 1600000;
constexpr int kIn    = 128;
constexpr int kHid   = 64;
constexpr int kOut   = 32;

typedef __attribute__((ext_vector_type(2))) float v2f;
typedef __attribute__((ext_vector_type(8))) float v8f;

// ---------------- degree / normalization ----------------

__global__ void zero_kernel(float* __restrict__ p, int n) {
    int i = blockIdx.x * blockDim.x + threadIdx.x;
    if (i < n) p[i] = 0.0f;
}

__global__ void degree_kernel(const int* __restrict__ dst, float* __restrict__ deg) {
    int e = blockIdx.x * blockDim.x + threadIdx.x;
    if (e < kEdges) atomicAdd(&deg[dst[e]], 1.0f);
}

__global__ void dinv_kernel(float* __restrict__ deg) {
    int i = blockIdx.x * blockDim.x + threadIdx.x;
    if (i < kNodes) deg[i] = rsqrtf(deg[i] + 1.0f);   // degree incl. self loop
}

// ---------------- fp32 WMMA GEMM + fused self-loop epilogue ----------------
// H[M,N] = X[M,K] @ W[K,N];  agg = dinv^2 * H   (all row-major)
// Block = 256 threads = 8 waves; each wave owns a 16-row strip and all N cols.
//
// LDS fragment layout: for p in [0, K/2), c in [0, N):
//   sW[(p*N + c)*2 + j] = W[(2p + j)*N + c],  j in {0,1}
// A lane (half h, col c, k-step k) reads its B pair as ONE float2 at
//   sW + ((k/2 + h)*N + c)*2        (ds_load_b64, operand-ready pair)

template <int K, int N>
__global__ void __launch_bounds__(256)
gemm_wmma(const float* __restrict__ X, const float* __restrict__ W,
          float* __restrict__ H, const float* __restrict__ dinv,
          float* __restrict__ agg) {
    constexpr int NT = N / 16;            // N tiles per wave
    __shared__ float sW[K * N];

    // stage weights into fragment layout (one-time, <= 32 KB)
    for (int idx = threadIdx.x; idx < (K / 2) * N; idx += blockDim.x) {
        int p = idx / N;
        int c = idx - p * N;
        v2f v;
        v.x = W[(2 * p) * N + c];
        v.y = W[(2 * p + 1) * N + c];
        *reinterpret_cast<v2f*>(sW + idx * 2) = v;
    }
    __syncthreads();

    const int wave = threadIdx.x >> 5;
    const int lane = threadIdx.x & 31;
    const int half = lane >> 4;           // 0: lanes 0-15, 1: lanes 16-31
    const int l15  = lane & 15;
    const int m0   = (blockIdx.x * 8 + wave) * 16;
    if (m0 >= kNodes) return;             // tail waves (after barrier)

    // A fragment source: lane needs X[m0+l15][k + 2*half .. +1] -> float2
    const float* ap = X + (size_t)(m0 + l15) * K + 2 * half;
    // B fragment base in LDS (immediate offsets added per k/tile)
    const float* bp = sW + ((size_t)half * N + l15) * 2;

    v8f acc[NT];
#pragma unroll
    for (int t = 0; t < NT; ++t) acc[t] = (v8f){};

#pragma unroll
    for (int k = 0; k < K; k += 4) {
        v2f a = *reinterpret_cast<const v2f*>(ap + k);     // global_load_b64
#pragma unroll
        for (int t = 0; t < NT; ++t) {
            v2f b = *reinterpret_cast<const v2f*>(bp + k * N + t * 32); // ds_load_b64
            acc[t] = __builtin_amdgcn_wmma_f32_16x16x4_f32(
                false, a, false, b, (short)0, acc[t], false, false);
        }
    }

    // Epilogue: store h and agg = dinv^2 * h.
    // C/D layout: VGPR j -> row m0 + half*8 + j, col = tile*16 + l15.
    float d2[8];
#pragma unroll
    for (int j = 0; j < 8; ++j) {
        float dv = dinv[m0 + half * 8 + j];
        d2[j] = dv * dv;
    }
#pragma unroll
    for (int t = 0; t < NT; ++t) {
        float* ho = H   + (size_t)(m0 + half * 8) * N + t * 16 + l15;
        float* ao = agg + (size_t)(m0 + half * 8) * N + t * 16 + l15;
#pragma unroll
        for (int j = 0; j < 8; ++j) {
            float v = acc[t][j];
            ho[(size_t)j * N] = v;
            ao[(size_t)j * N] = d2[j] * v;
        }
    }
}

// ---------------- edge scatter: agg[dst] += dinv[src]*dinv[dst] * h[src] ----------------
// F/4 threads per edge; float4 gather + 4 float atomics (L2-resident RMW).

template <int F>
__global__ void edge_kernel(const int* __restrict__ src, const int* __restrict__ dst,
                            const float* __restrict__ dinv, const float* __restrict__ h,
                            float* __restrict__ agg) {
    constexpr int TPE = F / 4;
    int gid = blockIdx.x * blockDim.x + threadIdx.x;
    int e = gid / TPE;                     // TPE is a power of two -> shift
    if (e >= kEdges) return;
    int c = (gid - e * TPE) * 4;
    int s = src[e];
    int d = dst[e];
    float norm = dinv[s] * dinv[d];
    float4 hv = *reinterpret_cast<const float4*>(h + (size_t)s * F + c);
    float* ap = agg + (size_t)d * F + c;
    atomicAdd(ap + 0, hv.x * norm);
    atomicAdd(ap + 1, hv.y * norm);
    atomicAdd(ap + 2, hv.z * norm);
    atomicAdd(ap + 3, hv.w * norm);
}

// ---------------- bias (+ optional ReLU), float4 ----------------

template <int F, bool RELU>
__global__ void bias_act_kernel(const float* __restrict__ agg,
                                const float* __restrict__ bias,
                                float* __restrict__ out) {
    constexpr int FQ = F / 4;
    int idx = blockIdx.x * blockDim.x + threadIdx.x;
    if (idx >= kNodes * FQ) return;
    int node = idx / FQ;
    int c    = (idx - node * FQ) * 4;
    float4 v = *reinterpret_cast<const float4*>(agg + (size_t)node * F + c);
    float4 b = *reinterpret_cast<const float4*>(bias + c);
    float4 r = make_float4(v.x + b.x, v.y + b.y, v.z + b.z, v.w + b.w);
    if (RELU) {
        r.x = fmaxf(r.x, 0.0f); r.y = fmaxf(r.y, 0.0f);
        r.z = fmaxf(r.z, 0.0f); r.w = fmaxf(r.w, 0.0f);
    }
    *reinterpret_cast<float4*>(out + (size_t)node * F + c) = r;
}

// ---------------- launch ----------------

extern "C" void kernel_launch(void* const* d_in, const int* in_sizes, int n_in,
                              void* d_out, int out_size, void* d_ws, size_t ws_size,
                              hipStream_t stream) {
    (void)in_sizes; (void)n_in; (void)out_size; (void)ws_size;

    const float* x  = (const float*)d_in[0];
    const int*   ei = (const int*)d_in[1];       // [2, E]: row 0 = src, row 1 = dst
    const float* W1 = (const float*)d_in[2];
    const float* b1 = (const float*)d_in[3];
    const float* W2 = (const float*)d_in[4];
    const float* b2 = (const float*)d_in[5];
    const float* W3 = (const float*)d_in[6];
    const float* b3 = (const float*)d_in[7];
    float* out = (float*)d_out;

    // workspace layout (floats): dinv | h | agg | xin   (~77 MB)
    float* dinv = (float*)d_ws;
    float* h    = dinv + kNodes;
    float* agg  = h   + (size_t)kNodes * kHid;
    float* xin  = agg + (size_t)kNodes * kHid;

    const int* src = ei;
    const int* dst = ei + kEdges;

    const int B = 256;
    const int nodeBlk = (kNodes + B - 1) / B;
    const int edgeBlk = (kEdges + B - 1) / B;
    const int vecHid  = (kNodes * (kHid / 4) + B - 1) / B;
    const int vecOut  = (kNodes * (kOut / 4) + B - 1) / B;
    const int edgeHid = (int)(((size_t)kEdges * (kHid / 4) + B - 1) / B);
    const int edgeOut = (int)(((size_t)kEdges * (kOut / 4) + B - 1) / B);
    const int gemmBlk = (kNodes / 16 + 7) / 8;   // 8 waves/block, 16 rows/wave

    // degrees -> dinv
    zero_kernel  <<<nodeBlk, B, 0, stream>>>(dinv, kNodes);
    degree_kernel<<<edgeBlk, B, 0, stream>>>(dst, dinv);
    dinv_kernel  <<<nodeBlk, B, 0, stream>>>(dinv);

    // ---- layer 1: x[128] -> h[64], ReLU ----
    gemm_wmma<kIn, kHid><<<gemmBlk, B, 0, stream>>>(x, W1, h, dinv, agg);
    edge_kernel<kHid><<<edgeHid, B, 0, stream>>>(src, dst, dinv, h, agg);
    bias_act_kernel<kHid, true><<<vecHid, B, 0, stream>>>(agg, b1, xin);

    // ---- layer 2: xin[64] -> h[64], ReLU ----
    gemm_wmma<kHid, kHid><<<gemmBlk, B, 0, stream>>>(xin, W2, h, dinv, agg);
    edge_kernel<kHid><<<edgeHid, B, 0, stream>>>(src, dst, dinv, h, agg);
    bias_act_kernel<kHid, true><<<vecHid, B, 0, stream>>>(agg, b2, xin);

    // ---- layer 3: xin[64] -> out[32], no ReLU, accumulate into d_out ----
    gemm_wmma<kHid, kOut><<<gemmBlk, B, 0, stream>>>(xin, W3, h, dinv, out);
    edge_kernel<kOut><<<edgeOut, B, 0, stream>>>(src, dst, dinv, h, out);
    bias_act_kernel<kOut, false><<<vecOut, B, 0, stream>>>(out, b3, out);
}